// OutputLayer_48670569398563
// MI455X (gfx1250) — compile-verified
//
#include <hip/hip_runtime.h>

typedef int   v4i __attribute__((ext_vector_type(4)));
typedef float v4f __attribute__((ext_vector_type(4)));

// Non-returning, relaxed, device-scope f32 atomic add -> global_atomic_add_f32
__device__ __forceinline__ void atomic_add_noret(float* p, float v) {
    (void)__hip_atomic_fetch_add(p, v, __ATOMIC_RELAXED, __HIP_MEMORY_SCOPE_AGENT);
}

// ---------------------------------------------------------------------------
// Tiny tail kernel FIRST so the disasm snippet shows the atomic codegen
// (same atomic path as the hot kernel). Never launched for E = 16M.
// ---------------------------------------------------------------------------
__global__ void scatter_tail_kernel(const float* __restrict__ pred,
                                    const int*   __restrict__ src,
                                    const int*   __restrict__ dst,
                                    float*       __restrict__ acc,
                                    int start, int e) {
    int i = start + (int)(blockIdx.x * blockDim.x + threadIdx.x);
    if (i < e) {
        atomic_add_noret(acc + dst[i], pred[src[i]]);
    }
}

// ---------------------------------------------------------------------------
// HOT KERNEL: edge scatter-add, 4 edges per thread.
//  - b128 TH_LOAD_NT loads for the once-streamed src/dst index arrays
//  - global_prefetch_b8 ~32KB ahead, WGP scope (fills all cache levels;
//    guarded so addresses are always valid -> non-speculative OK)
//  - regular-temporal gathers from the 4MB pred table (L2-resident)
//  - no-return global_atomic_add_f32 into the 4MB accumulator (L2-resident)
// ---------------------------------------------------------------------------
__global__ void scatter_kernel(const float* __restrict__ pred,
                               const int*   __restrict__ src,
                               const int*   __restrict__ dst,
                               float*       __restrict__ acc,
                               int e4) {
    int t = blockIdx.x * blockDim.x + threadIdx.x;
    if (t >= e4) return;

    if (t + 2048 < e4) {
        __builtin_prefetch((const void*)((const v4i*)src + t + 2048), 0, 3);
        __builtin_prefetch((const void*)((const v4i*)dst + t + 2048), 0, 3);
    }

    v4i s = __builtin_nontemporal_load((const v4i*)src + t);
    v4i d = __builtin_nontemporal_load((const v4i*)dst + t);

    float m0 = pred[s.x];
    float m1 = pred[s.y];
    float m2 = pred[s.z];
    float m3 = pred[s.w];

    atomic_add_noret(acc + d.x, m0);
    atomic_add_noret(acc + d.y, m1);
    atomic_add_noret(acc + d.z, m2);
    atomic_add_noret(acc + d.w, m3);
}

// ---------------------------------------------------------------------------
// Zero the accumulator (d_out doubles as the segment-sum accumulator).
// Regular-temporal stores: we WANT these lines L2-resident for the atomics.
// ---------------------------------------------------------------------------
__global__ void zero_kernel(float* __restrict__ acc, int n, int n4) {
    int i = blockIdx.x * blockDim.x + threadIdx.x;
    if (i >= n4) return;
    int base = i * 4;
    if (base + 3 < n) {
        v4f z = (v4f)0.0f;
        *((v4f*)(acc + base)) = z;
    } else {
        if (base     < n) acc[base]     = 0.0f;
        if (base + 1 < n) acc[base + 1] = 0.0f;
        if (base + 2 < n) acc[base + 2] = 0.0f;
    }
}

// ---------------------------------------------------------------------------
// In-place apply: out[i] = (sum[i] - 1)^2 * cost[i]
// ---------------------------------------------------------------------------
__global__ void apply_kernel(float* __restrict__ acc,
                             const float* __restrict__ cost,
                             int n, int n4) {
    int i = blockIdx.x * blockDim.x + threadIdx.x;
    if (i >= n4) return;
    int base = i * 4;
    if (base + 3 < n) {
        v4f s = *((const v4f*)(acc + base));
        v4f c = __builtin_nontemporal_load((const v4f*)(cost + base));
        v4f h = s - 1.0f;
        v4f r = h * h * c;
        __builtin_nontemporal_store(r, (v4f*)(acc + base));
    } else {
        if (base < n)     { float h = acc[base]     - 1.0f; acc[base]     = h * h * cost[base];     }
        if (base + 1 < n) { float h = acc[base + 1] - 1.0f; acc[base + 1] = h * h * cost[base + 1]; }
        if (base + 2 < n) { float h = acc[base + 2] - 1.0f; acc[base + 2] = h * h * cost[base + 2]; }
    }
}

extern "C" void kernel_launch(void* const* d_in, const int* in_sizes, int n_in,
                              void* d_out, int out_size, void* d_ws, size_t ws_size,
                              hipStream_t stream) {
    const float* pred = (const float*)d_in[0];   // pred_prob    [N] f32
    const float* cost = (const float*)d_in[1];   // special_cost [N] f32
    const int*   src  = (const int*)d_in[2];     // src [E] i32
    const int*   dst  = (const int*)d_in[3];     // dst [E] i32
    float*       out  = (float*)d_out;           // [N] f32 (doubles as accumulator)

    const int n   = out_size;        // num_nodes
    const int e   = in_sizes[2];     // num_edges
    const int n4  = (n + 3) / 4;
    const int e4  = e / 4;
    const int rem = e - e4 * 4;

    const int BLK = 256;  // 8 wave32 waves per block

    zero_kernel<<<(n4 + BLK - 1) / BLK, BLK, 0, stream>>>(out, n, n4);

    if (e4 > 0) {
        scatter_kernel<<<(e4 + BLK - 1) / BLK, BLK, 0, stream>>>(pred, src, dst, out, e4);
    }
    if (rem > 0) {
        scatter_tail_kernel<<<1, 32, 0, stream>>>(pred, src, dst, out, e4 * 4, e);
    }

    apply_kernel<<<(n4 + BLK - 1) / BLK, BLK, 0, stream>>>(out, cost, n, n4);
}